// LatentLongRange_27943057228064
// MI455X (gfx1250) — compile-verified
//
#include <hip/hip_runtime.h>
#include <hip/hip_bf16.h>

typedef float v2f __attribute__((ext_vector_type(2)));
typedef float v8f __attribute__((ext_vector_type(8)));

#define F_DIM      128
#define NUM_GRAPHS 48
#define KE_CONST   14.3996f
#define SQRT2_C    1.41421356f
#define PAIR_SPLIT 4

// ---------------------------------------------------------------------------
// Kernel 1: fused dual-head MLP via V_WMMA_F32_16X16X4_F32 (full f32 path).
// One block = 16 atom rows. 8 waves; wave w owns output column tile [16w,16w+16)
// for BOTH heads (shared A fragments, two accumulators -> 64 WMMAs per wave).
// After SiLU, tiles land in LDS and thin reductions produce:
//   q_raw[i]  = Hq[i,:] @ qW2
//   sc6[i]    = sqrt(softplus(Hv[i,:] @ vW2[:,0] + vb2[0]))
//   rvdw[i]   =      softplus(Hv[i,:] @ vW2[:,1] + vb2[1])
// ---------------------------------------------------------------------------
__global__ __launch_bounds__(256) void llr_mlp_wmma(
    const float* __restrict__ h0,
    const float* __restrict__ qW1, const float* __restrict__ qb1,
    const float* __restrict__ qW2,
    const float* __restrict__ vW1, const float* __restrict__ vb1,
    const float* __restrict__ vW2, const float* __restrict__ vb2,
    float* __restrict__ q_raw, float* __restrict__ sc6, float* __restrict__ rvdw)
{
    __shared__ float Hq[16 * F_DIM];
    __shared__ float Hv[16 * F_DIM];

    const int rowBase = blockIdx.x * 16;
    const int lane = threadIdx.x & 31;
    const int wave = threadIdx.x >> 5;   // 0..7 -> column tile
    const int n0   = wave * 16;
    const int hsel = lane >> 4;          // 0: K pair {0,1}; 1: K pair {2,3}
    const int l15  = lane & 15;

    const float* aRow = h0 + (size_t)(rowBase + l15) * F_DIM;

    v8f accq = {0.f, 0.f, 0.f, 0.f, 0.f, 0.f, 0.f, 0.f};
    v8f accv = {0.f, 0.f, 0.f, 0.f, 0.f, 0.f, 0.f, 0.f};

    #pragma unroll 4
    for (int kk = 0; kk < F_DIM / 4; ++kk) {
        const int k0 = kk * 4 + hsel * 2;
        v2f a, bq, bv;
        // A 16x4 f32 fragment: lanes 0-15 hold K={0,1}, lanes 16-31 K={2,3}
        a[0] = aRow[k0];
        a[1] = aRow[k0 + 1];
        // B 4x16 fragments (row K striped across lanes)
        bq[0] = qW1[(k0    ) * F_DIM + n0 + l15];
        bq[1] = qW1[(k0 + 1) * F_DIM + n0 + l15];
        bv[0] = vW1[(k0    ) * F_DIM + n0 + l15];
        bv[1] = vW1[(k0 + 1) * F_DIM + n0 + l15];
        accq = __builtin_amdgcn_wmma_f32_16x16x4_f32(false, a, false, bq,
                                                     (short)0, accq, false, false);
        accv = __builtin_amdgcn_wmma_f32_16x16x4_f32(false, a, false, bv,
                                                     (short)0, accv, false, false);
    }

    // bias + SiLU, scatter C fragment (VGPR v: row v / v+8, col = lane&15) to LDS
    const int col = n0 + l15;
    const float biasq = qb1[col];
    const float biasv = vb1[col];
    #pragma unroll
    for (int v = 0; v < 8; ++v) {
        const int r = v + (hsel << 3);
        float xq = accq[v] + biasq;
        float xv = accv[v] + biasv;
        Hq[r * F_DIM + col] = xq / (1.0f + __expf(-xq));
        Hv[r * F_DIM + col] = xv / (1.0f + __expf(-xv));
    }
    __syncthreads();

    const int t = threadIdx.x;
    if (t < 16) {                         // wave 0: q head (no bias on W2)
        float acc = 0.0f;
        for (int c = 0; c < F_DIM; ++c) acc += Hq[t * F_DIM + c] * qW2[c];
        q_raw[rowBase + t] = acc;
    } else if (t >= 32 && t < 48) {       // wave 1: c6 channel
        const int r = t - 32;
        float acc = vb2[0];
        for (int c = 0; c < F_DIM; ++c) acc += Hv[r * F_DIM + c] * vW2[2 * c];
        float sp = fmaxf(acc, 0.0f) + log1pf(expf(-fabsf(acc)));   // softplus
        sc6[rowBase + r] = sqrtf(sp);     // pre-sqrt so pairwise c6_ij = s_i*s_j
    } else if (t >= 64 && t < 80) {       // wave 2: r_vdw channel
        const int r = t - 64;
        float acc = vb2[1];
        for (int c = 0; c < F_DIM; ++c) acc += Hv[r * F_DIM + c] * vW2[2 * c + 1];
        rvdw[rowBase + r] = fmaxf(acc, 0.0f) + log1pf(expf(-fabsf(acc)));
    }
}

// ---------------------------------------------------------------------------
// Kernel 2: batch is sorted -> per-graph segment starts + per-graph q means.
// Thread g: segs[g] = #{i : batch[i] < g}  (g=0..48, segs[48]=N),
//           qmean[g] = segment mean of q_raw. Deterministic sequential scans.
// ---------------------------------------------------------------------------
__global__ __launch_bounds__(64) void llr_segments(
    const float* __restrict__ q_raw, const int* __restrict__ batch,
    float* __restrict__ qmean, int* __restrict__ segs, int N)
{
    const int g = threadIdx.x;
    if (g > NUM_GRAPHS) return;
    int lt = 0; float s = 0.0f; int c = 0;
    for (int i = 0; i < N; ++i) {
        const int b = batch[i];
        lt += (b < g) ? 1 : 0;
        if (g < NUM_GRAPHS && b == g) { s += q_raw[i]; c += 1; }
    }
    segs[g] = lt;
    if (g < NUM_GRAPHS) qmean[g] = s / fmaxf((float)c, 1.0f);
}

// Kernel 3: q centered per graph
__global__ __launch_bounds__(256) void llr_center(
    const float* __restrict__ q_raw, const int* __restrict__ batch,
    const float* __restrict__ qmean, float* __restrict__ qc, int N)
{
    const int i = blockIdx.x * blockDim.x + threadIdx.x;
    if (i < N) qc[i] = q_raw[i] - qmean[batch[i]];
}

// ---------------------------------------------------------------------------
// Kernel 4: segment-local pairwise energy (mask == intra-segment, j != i),
// 48x less work than the dense N^2 masked sum. PAIR_SPLIT lanes share one
// atom's j-loop. Fixed-order tree reduction -> one partial per block.
// Accumulates  KE*q_i*q_j*erf(d/(sqrt2*sigma))/(d+eps) - c6ij/(d2^3+(r_i r_j)^3+eps);
// the global 0.5 factor is applied in the finish kernel.
// ---------------------------------------------------------------------------
__global__ __launch_bounds__(256) void llr_pair(
    const float* __restrict__ pos, const float* __restrict__ qc,
    const float* __restrict__ sc6, const float* __restrict__ rvdw,
    const int* __restrict__ batch, const float* __restrict__ sigma,
    const int* __restrict__ segs, float* __restrict__ partial)
{
    __shared__ float red[256];
    const int t   = threadIdx.x;
    const int idx = blockIdx.x * 256 + t;
    const int i   = idx >> 2;             // PAIR_SPLIT == 4
    const int sl  = idx & 3;

    const float xi = pos[3 * i], yi = pos[3 * i + 1], zi = pos[3 * i + 2];
    const float qi = qc[i], si = sc6[i], ri = rvdw[i];
    const int   b  = batch[i];
    const int   js = segs[b], je = segs[b + 1];
    const float inv_s = 1.0f / (SQRT2_C * sigma[0]);

    float e = 0.0f;
    for (int j = js + sl; j < je; j += PAIR_SPLIT) {
        if (j == i) continue;
        const float dx = xi - pos[3 * j];
        const float dy = yi - pos[3 * j + 1];
        const float dz = zi - pos[3 * j + 2];
        const float d2 = dx * dx + dy * dy + dz * dz;
        const float dist = sqrtf(d2 + 1e-8f);
        const float elec = KE_CONST * qi * qc[j] * erff(dist * inv_s) / (dist + 1e-8f);
        const float pr = ri * rvdw[j];                       // (r_i*r_j)
        const float denom = d2 * d2 * d2 + pr * pr * pr + 1e-8f;
        e += elec - (si * sc6[j]) / denom;
    }

    red[t] = e;
    __syncthreads();
    for (int w = 128; w > 0; w >>= 1) {
        if (t < w) red[t] += red[t + w];
        __syncthreads();
    }
    if (t == 0) partial[blockIdx.x] = red[0];
}

// Kernel 5: deterministic final reduction, apply the global 0.5 factor.
__global__ __launch_bounds__(256) void llr_finish(
    const float* __restrict__ partial, int n, float* __restrict__ out)
{
    __shared__ float red[256];
    const int t = threadIdx.x;
    float s = 0.0f;
    for (int i = t; i < n; i += 256) s += partial[i];
    red[t] = s;
    __syncthreads();
    for (int w = 128; w > 0; w >>= 1) {
        if (t < w) red[t] += red[t + w];
        __syncthreads();
    }
    if (t == 0) out[0] = 0.5f * red[0];
}

extern "C" void kernel_launch(void* const* d_in, const int* in_sizes, int n_in,
                              void* d_out, int out_size, void* d_ws, size_t ws_size,
                              hipStream_t stream)
{
    const float* h0    = (const float*)d_in[0];
    // d_in[1] = h1 : accepted but unused on this path
    const float* pos   = (const float*)d_in[2];
    const float* qW1   = (const float*)d_in[3];
    const float* qb1   = (const float*)d_in[4];
    const float* qW2   = (const float*)d_in[5];
    const float* sigma = (const float*)d_in[6];
    const float* vW1   = (const float*)d_in[7];
    const float* vb1   = (const float*)d_in[8];
    const float* vW2   = (const float*)d_in[9];
    const float* vb2   = (const float*)d_in[10];
    const int*   batch = (const int*)d_in[11];

    const int N = in_sizes[0] / F_DIM;   // 6144

    // workspace layout (all 4-byte aligned)
    float* q_raw   = (float*)d_ws;
    float* qcv     = q_raw + N;
    float* sc6     = qcv + N;
    float* rvd     = sc6 + N;
    float* qmean   = rvd + N;                 // 64 slots
    int*   segs    = (int*)(qmean + 64);      // 64 slots (needs 49)
    float* partial = (float*)(segs + 64);

    llr_mlp_wmma<<<N / 16, 256, 0, stream>>>(h0, qW1, qb1, qW2, vW1, vb1, vW2, vb2,
                                             q_raw, sc6, rvd);
    llr_segments<<<1, 64, 0, stream>>>(q_raw, batch, qmean, segs, N);
    llr_center<<<(N + 255) / 256, 256, 0, stream>>>(q_raw, batch, qmean, qcv, N);

    const int npb = (N * PAIR_SPLIT) / 256;   // 96 blocks
    llr_pair<<<npb, 256, 0, stream>>>(pos, qcv, sc6, rvd, batch, sigma, segs, partial);
    llr_finish<<<1, 256, 0, stream>>>(partial, npb, (float*)d_out);
}